// TemporalCooccurrenceMatrix_22539988370104
// MI455X (gfx1250) — compile-verified
//
#include <hip/hip_runtime.h>
#include <hip/hip_bf16.h>

// CDNA5 / gfx1250 — wave32, WMMA.
typedef __attribute__((ext_vector_type(2))) float v2f;
typedef __attribute__((ext_vector_type(8))) float v8f;

#define NB   8      // batches
#define NW   128    // walks
#define LW   20     // walk length
#define TILE 16     // 16x16 walk-pair tile per workgroup
#define QR   (TILE * LW)   // 320 rows of Q
#define QS   17            // padded row stride (floats) to dodge bank conflicts
#define DKSTRIDE 50        // dist_kernel is 50x50

__global__ __launch_bounds__(256)
void TemporalCooccurrenceMatrix_22539988370104_kernel(
    const int* __restrict__ nodes,          // (B, W, L) int32
    const unsigned char* __restrict__ masks,// (B, W, L) bool (1 byte)
    const float* __restrict__ times,        // (B, W, L) f32
    const float* __restrict__ distk,        // (50, 50) f32
    float* __restrict__ out)                // (B, W, W) f32
{
    __shared__ float Q[QR][QS];     // Q[(w1l*20+l1)][w2l] = sum_l2 weight
    __shared__ float Dk[LW][LW];    // dist kernel 20x20 slice
    __shared__ int   n1[QR]; __shared__ float t1s[QR]; __shared__ float m1s[QR];
    __shared__ int   n2[QR]; __shared__ float t2s[QR]; __shared__ float m2s[QR];
    __shared__ float Cacc[TILE * TILE];  // merged 16x16 tile accumulator

    const int tid = threadIdx.x;
    const int b   = blockIdx.z;
    const int w1b = blockIdx.y * TILE;
    const int w2b = blockIdx.x * TILE;
    const int g1  = (b * NW + w1b) * LW;   // 320 contiguous elements
    const int g2  = (b * NW + w2b) * LW;

    // ---- Stage walk data + dist kernel into LDS ----
    for (int i = tid; i < QR; i += 256) {
        n1[i]  = nodes[g1 + i];
        t1s[i] = times[g1 + i];
        m1s[i] = masks[g1 + i] ? 1.0f : 0.0f;
        n2[i]  = nodes[g2 + i];
        t2s[i] = times[g2 + i];
        m2s[i] = masks[g2 + i] ? 1.0f : 0.0f;
    }
    for (int i = tid; i < LW * LW; i += 256) {
        Dk[i / LW][i % LW] = distk[(i / LW) * DKSTRIDE + (i % LW)];
    }
    Cacc[tid] = 0.0f;
    __syncthreads();

    // ---- Phase 1 (VALU): pairwise gated weights, reduced over l2 ----
    // weight = D[l1,l2] * exp(-min(|dt|,1e4)/5); clip(.,0,1) is a no-op since
    // both factors lie in (0,1]. Gate by (same node) && (both masks).
    for (int e = tid; e < QR * TILE; e += 256) {
        const int r  = e >> 4;      // row: w1_local*20 + l1
        const int c  = e & 15;      // col: w2_local
        const int l1 = r % LW;
        const int   nn = n1[r];
        const float tt = t1s[r];
        const float mm = m1s[r];
        const int base2 = c * LW;
        float acc = 0.0f;
#pragma unroll
        for (int l2 = 0; l2 < LW; ++l2) {
            const int j = base2 + l2;
            const float sel = (nn == n2[j]) ? (mm * m2s[j]) : 0.0f;
            float dt = fminf(fabsf(tt - t2s[j]), 10000.0f);
            acc = fmaf(sel * Dk[l1][l2], __expf(dt * -0.2f), acc);
        }
        Q[r][c] = acc;
    }
    __syncthreads();

    // ---- Phase 2 (WMMA): C_tile = A_indicator(16x320) x Q(320x16) ----
    // A[m,k] = (k/20 == m). K split: wave w owns steps [10w, 10w+10), i.e.
    // K in [40w, 40w+40). 20 % 4 == 0 so each K-quad stays in one walk, and
    // each wave only ever sees two indicator rows: 2w (steps 0-4) and 2w+1
    // (steps 5-9) — hoist both A operands out of the loop so the WMMA chain
    // has no per-step integer-division dependency.
    // Layouts (ISA 7.12.2):
    //   A 16x4 f32: lane M = lane%16; VGPR0 -> K = (lane<16?0:2), VGPR1 -> K+1.
    //   B 4x16 f32: N = lane%16;      VGPR0 -> K = (lane<16?0:2), VGPR1 -> K+1.
    //   C 16x16 f32: M = v + 8*(lane/16), N = lane%16.
    const int wave  = tid >> 5;
    const int lane  = tid & 31;
    const int half  = lane >> 4;    // 0: lanes 0-15, 1: lanes 16-31
    const int lmod  = lane & 15;

    const float av0 = (lmod == 2 * wave)     ? 1.0f : 0.0f;
    const float av1 = (lmod == 2 * wave + 1) ? 1.0f : 0.0f;
    v2f a_lo; a_lo.x = av0; a_lo.y = av0;   // all 4 K's of a quad hit one walk
    v2f a_hi; a_hi.x = av1; a_hi.y = av1;

    const int kbase = wave * 40;            // wave's K window start
    const int roff0 = half ? 2 : 0;         // VGPR0 K slot for this lane
    const int roff1 = half ? 3 : 1;         // VGPR1 K slot for this lane

    v8f c8 = {};
#pragma unroll
    for (int s = 0; s < 5; ++s) {
        const int k0 = kbase + s * 4;
        v2f bv;
        bv.x = Q[k0 + roff0][lmod];
        bv.y = Q[k0 + roff1][lmod];
        // 8 args: (neg_a, A, neg_b, B, c_mod, C, reuse_a, reuse_b)
        c8 = __builtin_amdgcn_wmma_f32_16x16x4_f32(
            false, a_lo, false, bv, (short)0, c8, false, false);
    }
#pragma unroll
    for (int s = 5; s < 10; ++s) {
        const int k0 = kbase + s * 4;
        v2f bv;
        bv.x = Q[k0 + roff0][lmod];
        bv.y = Q[k0 + roff1][lmod];
        c8 = __builtin_amdgcn_wmma_f32_16x16x4_f32(
            false, a_hi, false, bv, (short)0, c8, false, false);
    }

    // Merge the 8 per-wave partial accumulators via LDS float atomics
    // (ds_add_f32), then finish with tanh(clip(.,-10,10)).
#pragma unroll
    for (int v = 0; v < 8; ++v) {
        const int M = v + 8 * half;
        atomicAdd(&Cacc[M * TILE + lmod], c8[v]);
    }
    __syncthreads();

    const int m = tid >> 4;
    const int n = tid & 15;
    const float s  = Cacc[tid];    // tid == m*16 + n == M*16 + N
    const float o  = tanhf(fminf(fmaxf(s, -10.0f), 10.0f));
    out[(b * NW + (w1b + m)) * NW + (w2b + n)] = o;
}

extern "C" void kernel_launch(void* const* d_in, const int* in_sizes, int n_in,
                              void* d_out, int out_size, void* d_ws, size_t ws_size,
                              hipStream_t stream) {
    (void)in_sizes; (void)n_in; (void)d_ws; (void)ws_size; (void)out_size;
    const int*           nodes = (const int*)d_in[0];           // int32 (x64 off)
    const unsigned char* masks = (const unsigned char*)d_in[1]; // bool, 1 byte
    const float*         times = (const float*)d_in[2];
    const float*         distk = (const float*)d_in[3];         // 50x50
    float*               out   = (float*)d_out;                 // (8,128,128)

    dim3 grid(NW / TILE, NW / TILE, NB);   // 8 x 8 x 8 = 512 workgroups
    dim3 block(256);                       // 8 wave32 waves
    TemporalCooccurrenceMatrix_22539988370104_kernel<<<grid, block, 0, stream>>>(
        nodes, masks, times, distk, out);
}